// GNNModule_5557687681129
// MI455X (gfx1250) — compile-verified
//
#include <hip/hip_runtime.h>

typedef float v2f __attribute__((ext_vector_type(2)));
typedef float v8f __attribute__((ext_vector_type(8)));

#define FDIM 32
#define KNB  16
#define LDP  33          // padded LDS row stride (bank-conflict free: gcd(33,64)=1)
#define MATS (16*LDP)    // one staged 16x32 tile
#define WPAD (32*LDP)    // one staged 32x32 weight matrix
#define WAVES 4

__device__ __forceinline__ void st_row16(float* d, float4 a, float4 b, float4 c, float4 e) {
  d[0]=a.x; d[1]=a.y; d[2]=a.z; d[3]=a.w;
  d[4]=b.x; d[5]=b.y; d[6]=b.z; d[7]=b.w;
  d[8]=c.x; d[9]=c.y; d[10]=c.z; d[11]=c.w;
  d[12]=e.x; d[13]=e.y; d[14]=e.z; d[15]=e.w;
}

// Fused: gather-sum neighbors + 5x (16x32)@(32x32) GEMM via f32 WMMA + half-ReLU
// + pre-BN store + hierarchical BN-stat accumulation.
__global__ void __launch_bounds__(128)
lgnn_side_kernel(const float* __restrict__ src, const float* __restrict__ deg,
                 const int* __restrict__ t, const int* __restrict__ tt,
                 const float* __restrict__ aux, const int* __restrict__ aux_idx,
                 const float* __restrict__ Wself, const float* __restrict__ Wdeg,
                 const float* __restrict__ Wr,   const float* __restrict__ Waux,
                 const float* __restrict__ bself, const float* __restrict__ bdeg,
                 const float* __restrict__ br,    const float* __restrict__ baux,
                 float* __restrict__ out_pre, double* __restrict__ stats, int M)
{
  __shared__ float  wsW[5*WPAD];             // 5 weight matrices, [k][n] padded
  __shared__ float  stage[WAVES][4*MATS];    // per-wave: self, t-sum, tt-sum, aux
  __shared__ float  sdeg[WAVES][16];
  __shared__ double ssum[FDIM];
  __shared__ double ssqs[FDIM];

  const int tid  = threadIdx.x;
  const int lane = tid & 31;
  const int wv   = tid >> 5;

  if (tid < FDIM) { ssum[tid] = 0.0; ssqs[tid] = 0.0; }

  { // cooperative weight staging (w is loop-constant -> no private-array indexing)
    const float* Ws[5] = {Wself, Wdeg, Wr, Wr + 1024, Waux};
#pragma unroll
    for (int w = 0; w < 5; ++w) {
      const float* Wp = Ws[w];
      for (int i = tid; i < 1024; i += 128)
        wsW[w*WPAD + (i >> 5)*LDP + (i & 31)] = Wp[i];
    }
  }

  const int tile   = blockIdx.x * WAVES + wv;
  const int ntiles = M >> 4;                 // M is a multiple of 16 here
  const bool valid = tile < ntiles;
  const int  r0    = valid ? tile * 16 : 0;  // clamp: keep EXEC all-1s for WMMA

  { // gather phase: 2 lanes per row, 16 columns each, float4 vector loads
    const int    rrow = lane >> 1;
    const int    ch   = (lane & 1) * 16;
    const size_t row  = (size_t)(r0 + rrow);
    float4 a0={0,0,0,0},a1={0,0,0,0},a2={0,0,0,0},a3={0,0,0,0};
    float4 b0={0,0,0,0},b1={0,0,0,0},b2={0,0,0,0},b3={0,0,0,0};
    const int* tp  = t  + row*KNB;
    const int* ttp = tt + row*KNB;
#pragma unroll 4
    for (int k = 0; k < KNB; ++k) {
      const float4* p = (const float4*)(src + (size_t)tp[k]*FDIM + ch);
      a0 += p[0]; a1 += p[1]; a2 += p[2]; a3 += p[3];
      const float4* q = (const float4*)(src + (size_t)ttp[k]*FDIM + ch);
      b0 += q[0]; b1 += q[1]; b2 += q[2]; b3 += q[3];
    }
    float* st = stage[wv];
    const float4* xr = (const float4*)(src + row*FDIM + ch);
    st_row16(st + 0*MATS + rrow*LDP + ch, xr[0], xr[1], xr[2], xr[3]);
    st_row16(st + 1*MATS + rrow*LDP + ch, a0, a1, a2, a3);
    st_row16(st + 2*MATS + rrow*LDP + ch, b0, b1, b2, b3);
    const size_t arow = aux_idx ? (size_t)aux_idx[row] : row;
    const float4* pr  = (const float4*)(aux + arow*FDIM + ch);
    st_row16(st + 3*MATS + rrow*LDP + ch, pr[0], pr[1], pr[2], pr[3]);
    if ((lane & 1) == 0) sdeg[wv][rrow] = deg[row];
  }

  __syncthreads();

  const float* lsx  = stage[wv] + 0*MATS;
  const float* lst  = stage[wv] + 1*MATS;
  const float* lstt = stage[wv] + 2*MATS;
  const float* lsa  = stage[wv] + 3*MATS;
  const int   m  = lane & 15;
  const int   hi = lane >> 4;               // A: lanes 16-31 hold K+2,K+3
  const float dm = sdeg[wv][m];
  const float* W0 = wsW + 0*WPAD;
  const float* W1 = wsW + 1*WPAD;
  const float* W2 = wsW + 2*WPAD;
  const float* W3 = wsW + 3*WPAD;
  const float* W4 = wsW + 4*WPAD;

#pragma unroll
  for (int nh = 0; nh < 2; ++nh) {          // two 16-col output halves
    const int   c    = nh*16 + m;           // this lane's output column
    const float bsum = bself[c] + bdeg[c] + br[c] + br[FDIM + c] + baux[c];
    v8f acc;
#pragma unroll
    for (int j = 0; j < 8; ++j) acc[j] = bsum;   // bias replicated over rows
#pragma unroll
    for (int kc = 0; kc < 8; ++kc) {        // K = 32 in steps of 4
      const int ka = kc*4 + 2*hi;
      v2f a, b;
      // self @ Wself
      a[0] = lsx[m*LDP + ka];  a[1] = lsx[m*LDP + ka + 1];
      b[0] = W0[ka*LDP + c];   b[1] = W0[(ka+1)*LDP + c];
      acc = __builtin_amdgcn_wmma_f32_16x16x4_f32(false, a, false, b, (short)0, acc, false, false);
      // (deg*self) @ Wdeg  (row scaling folded into A operand)
      v2f ad; ad[0] = a[0]*dm; ad[1] = a[1]*dm;
      b[0] = W1[ka*LDP + c];   b[1] = W1[(ka+1)*LDP + c];
      acc = __builtin_amdgcn_wmma_f32_16x16x4_f32(false, ad, false, b, (short)0, acc, false, false);
      // radius-0 aggregate @ Wr[0]
      a[0] = lst[m*LDP + ka];  a[1] = lst[m*LDP + ka + 1];
      b[0] = W2[ka*LDP + c];   b[1] = W2[(ka+1)*LDP + c];
      acc = __builtin_amdgcn_wmma_f32_16x16x4_f32(false, a, false, b, (short)0, acc, false, false);
      // radius-1 aggregate @ Wr[1]
      a[0] = lstt[m*LDP + ka]; a[1] = lstt[m*LDP + ka + 1];
      b[0] = W3[ka*LDP + c];   b[1] = W3[(ka+1)*LDP + c];
      acc = __builtin_amdgcn_wmma_f32_16x16x4_f32(false, a, false, b, (short)0, acc, false, false);
      // paired matrix @ Waux
      a[0] = lsa[m*LDP + ka];  a[1] = lsa[m*LDP + ka + 1];
      b[0] = W4[ka*LDP + c];   b[1] = W4[(ka+1)*LDP + c];
      acc = __builtin_amdgcn_wmma_f32_16x16x4_f32(false, a, false, b, (short)0, acc, false, false);
    }
    if (valid) {
      float s = 0.f, q = 0.f;
#pragma unroll
      for (int j = 0; j < 8; ++j) {         // rows M=j (lanes<16) / M=j+8 (lanes>=16)
        float v = acc[j];
        if (nh == 1) v = fmaxf(v, 0.f);     // ReLU on cols 16..31
        out_pre[(size_t)(r0 + j + 8*hi)*FDIM + c] = v;
        s += v; q += v*v;
      }
      atomicAdd(&ssum[c], (double)s);
      atomicAdd(&ssqs[c], (double)q);
    }
  }

  __syncthreads();
  if (tid < FDIM) {                          // one global f64 atomic per column/block
    atomicAdd(stats + tid,        ssum[tid]);
    atomicAdd(stats + FDIM + tid, ssqs[tid]);
  }
}

__global__ void init_ws(float* __restrict__ pmpd, long long nf,
                        double* __restrict__ stats, int nd) {
  long long i = (long long)blockIdx.x * blockDim.x + threadIdx.x;
  const long long stride = (long long)gridDim.x * blockDim.x;
  for (long long j = i; j < nf; j += stride) pmpd[j] = 0.f;
  if (i < nd) stats[i] = 0.0;
}

__global__ void scatter_segsum(const float* __restrict__ y, const int* __restrict__ dst,
                               float* __restrict__ pmpd, long long n) {
  long long i = (long long)blockIdx.x * blockDim.x + threadIdx.x;
  const long long stride = (long long)gridDim.x * blockDim.x;
  for (; i < n; i += stride) {
    const long long e = i >> 5;
    const int c = (int)(i & 31);
    atomicAdd(pmpd + (size_t)dst[e]*FDIM + c, y[i]);
  }
}

__global__ void bn_apply(float* __restrict__ data, const double* __restrict__ stats,
                         const float* __restrict__ w, const float* __restrict__ b,
                         long long rows) {
  const double inv = 1.0 / (double)rows;
  const long long n = rows * FDIM;
  long long i = (long long)blockIdx.x * blockDim.x + threadIdx.x;
  const long long stride = (long long)gridDim.x * blockDim.x;
  for (; i < n; i += stride) {
    const int c = (int)(i & 31);
    const double mean = stats[c] * inv;
    const double var  = stats[FDIM + c] * inv - mean * mean;
    const float  sc   = w[c] * rsqrtf((float)var + 1e-5f);
    data[i] = (float)((double)data[i] - mean) * sc + b[c];
  }
}

extern "C" void kernel_launch(void* const* d_in, const int* in_sizes, int n_in,
                              void* d_out, int out_size, void* d_ws, size_t ws_size,
                              hipStream_t stream) {
  const float* x       = (const float*)d_in[0];
  const float* y       = (const float*)d_in[1];
  const float* deg_g   = (const float*)d_in[2];
  const float* deg_lg  = (const float*)d_in[3];
  const int*   t_g     = (const int*)d_in[4];
  const int*   tt_g    = (const int*)d_in[5];
  const int*   t_lg    = (const int*)d_in[6];
  const int*   tt_lg   = (const int*)d_in[7];
  const int*   dst     = (const int*)d_in[8];
  const int*   pm_pd   = (const int*)d_in[9];
  const float* theta_x_w   = (const float*)d_in[10];
  const float* theta_x_b   = (const float*)d_in[11];
  const float* theta_deg_w = (const float*)d_in[12];
  const float* theta_deg_b = (const float*)d_in[13];
  const float* theta_y_w   = (const float*)d_in[14];
  const float* theta_y_b   = (const float*)d_in[15];
  const float* gamma_y_w   = (const float*)d_in[16];
  const float* gamma_y_b   = (const float*)d_in[17];
  const float* gamma_deg_w = (const float*)d_in[18];
  const float* gamma_deg_b = (const float*)d_in[19];
  const float* gamma_x_w   = (const float*)d_in[20];
  const float* gamma_x_b   = (const float*)d_in[21];
  const float* theta_r_w   = (const float*)d_in[22];
  const float* theta_r_b   = (const float*)d_in[23];
  const float* gamma_r_w   = (const float*)d_in[24];
  const float* gamma_r_b   = (const float*)d_in[25];
  const float* bn_x_w = (const float*)d_in[26];
  const float* bn_x_b = (const float*)d_in[27];
  const float* bn_y_w = (const float*)d_in[28];
  const float* bn_y_b = (const float*)d_in[29];

  const int N = in_sizes[0] / FDIM;
  const int E = in_sizes[1] / FDIM;

  float* out_x = (float*)d_out;
  float* out_y = out_x + (size_t)N * FDIM;

  float*  pmpd   = (float*)d_ws;                                  // N*32 floats
  double* stats  = (double*)((char*)d_ws + (size_t)N * FDIM * 4); // 128 doubles
  double* stat_x = stats;
  double* stat_y = stats + 2 * FDIM;

  // 1) zero segment-sum buffer + stats
  init_ws<<<2048, 256, 0, stream>>>(pmpd, (long long)N * FDIM, stats, 4 * FDIM);

  // 2) pmpd_y = segment_sum(y, dst, N)
  scatter_segsum<<<2048, 256, 0, stream>>>(y, dst, pmpd, (long long)E * FDIM);

  // 3) node side (x): aux = pmpd_y (identity indexing)
  {
    const int ntiles = N / 16;
    const int grid   = (ntiles + WAVES - 1) / WAVES;
    lgnn_side_kernel<<<grid, 128, 0, stream>>>(
        x, deg_g, t_g, tt_g, pmpd, (const int*)nullptr,
        theta_x_w, theta_deg_w, theta_r_w, theta_y_w,
        theta_x_b, theta_deg_b, theta_r_b, theta_y_b,
        out_x, stat_x, N);
  }

  // 4) edge side (y): aux = x gathered by pm_pd
  {
    const int ntiles = E / 16;
    const int grid   = (ntiles + WAVES - 1) / WAVES;
    lgnn_side_kernel<<<grid, 128, 0, stream>>>(
        y, deg_lg, t_lg, tt_lg, x, pm_pd,
        gamma_y_w, gamma_deg_w, gamma_r_w, gamma_x_w,
        gamma_y_b, gamma_deg_b, gamma_r_b, gamma_x_b,
        out_y, stat_y, E);
  }

  // 5) BatchNorm (training mode, biased stats), in place on d_out
  bn_apply<<<2048, 256, 0, stream>>>(out_x, stat_x, bn_x_w, bn_x_b, (long long)N);
  bn_apply<<<2048, 256, 0, stream>>>(out_y, stat_y, bn_y_w, bn_y_b, (long long)E);
}